// Canny_11871289606762
// MI455X (gfx1250) — compile-verified
//
#include <hip/hip_runtime.h>
#include <math.h>

// -------- CDNA5 wave32 WMMA types --------
typedef __attribute__((ext_vector_type(2))) float v2f;
typedef __attribute__((ext_vector_type(8))) float v8f;

#define HH 2048
#define WW 2048
#define HWSZ (HH * WW)

// LDS tile geometry: 64x64 outputs, 4-pixel halo -> 72x72 active region.
// Row stride 76: 76 mod 64 = 12 -> 12*L distinct mod 64 for L=0..15,
// so row-per-lane (WMMA A operand) LDS reads are bank-conflict free.
#define TROWS 72
#define TPW   76
#define LSLAB (TROWS * TPW)

__global__ __launch_bounds__(256)
void canny_fused_kernel(const float* __restrict__ img,
                        const float* __restrict__ gauss,
                        float* __restrict__ out)
{
    // Ping-pong LDS slabs: sA: input -> (pass2) blur ; sB: hblur -> (pass3) grad/gx/gy
    __shared__ float sA[3 * LSLAB];
    __shared__ float sB[3 * LSLAB];
    __shared__ float gLUT[64];   // zero-padded gaussian band LUT: gLUT[16+t] = g[t], t in [0,5)

    const int tid  = threadIdx.x;
    const int lane = tid & 31;
    const int wave = tid >> 5;        // 8 waves per block (wave32)
    const int j    = lane & 15;       // M (A) / N (B,D) index within half-wave
    const int half = lane >> 4;       // 0: K pair {0,1}, 1: K pair {2,3}
    const int gx0  = blockIdx.x * 64;
    const int gy0  = blockIdx.y * 64;

    // Gaussian band LUT (zero outside [0,5))
    if (tid < 64) {
        int t = tid - 16;
        gLUT[tid] = (t >= 0 && t < 5) ? gauss[t] : 0.0f;
    }

    // ---------- Pass 0: cooperative global -> LDS load of 72x72x3 halo tile ----------
    // gx0 is a multiple of 64, so (gx0-4) is 16B aligned and every float4 is
    // fully inside or fully outside the image (zero pad matches conv padding).
    for (int i = tid; i < 3 * 72 * 18; i += 256) {
        int ch  = i / (72 * 18);
        int rem = i - ch * (72 * 18);
        int r   = rem / 18;
        int c4  = (rem - r * 18) * 4;
        int gyy = gy0 - 4 + r;
        int gxx = gx0 - 4 + c4;
        float4 v = make_float4(0.f, 0.f, 0.f, 0.f);
        if (gyy >= 0 && gyy < HH && gxx >= 0 && gxx < WW)
            v = *(const float4*)(img + (size_t)ch * HWSZ + (size_t)gyy * WW + gxx);
        *(float4*)(&sA[ch * LSLAB + r * TPW + c4]) = v;
    }
    __syncthreads();

    // ---------- Constant banded-Gauss WMMA operand (identical for both passes) ----------
    // Horizontal pass:  B[c][j] = g[c - j]   (B operand, 20x16 band, K-sliced by 4)
    // Vertical pass:    A[i][c] = g[c - i]   (A operand, 16x20 band) -> same lane values.
    v2f band[5];
#pragma unroll
    for (int s = 0; s < 5; ++s) {
        int base = 4 * s + 2 * half - j + 16;   // in [1, 34]
        band[s].x = gLUT[base];
        band[s].y = gLUT[base + 1];
    }

    // Tile bases (last tile clamped so all accesses stay inside 72x76 slabs;
    // overlapped tiles recompute identical values -> benign).
    // Pass1 D rows: {0,16,32,48,56}; output cols / pass2 rows+cols: {2,18,34,50,54}.

    // ---------- Pass 1: horizontal Gaussian, stencil-as-GEMM via V_WMMA_F32_16X16X4_F32 ----------
    for (int t = wave; t < 75; t += 8) {          // uniform per wave: EXEC all-1s for WMMA
        int ch = t / 25;
        int r  = t - ch * 25;
        int ty = r / 5, tx = r - ty * 5;
        int yb = (ty < 4) ? 16 * ty : 56;
        int x0 = (tx < 4) ? 2 + 16 * tx : 54;
        const float* Ain = &sA[ch * LSLAB];
        v8f acc = {};
#pragma unroll
        for (int s = 0; s < 5; ++s) {
            int row = yb + j;
            int col = (x0 - 2) + 4 * s + 2 * half;   // <= 71
            v2f a;
            a.x = Ain[row * TPW + col];
            a.y = Ain[row * TPW + col + 1];
            acc = __builtin_amdgcn_wmma_f32_16x16x4_f32(
                      false, a, false, band[s], (short)0, acc, false, false);
        }
        float* Bout = &sB[ch * LSLAB];
#pragma unroll
        for (int r8 = 0; r8 < 8; ++r8)            // D layout: M = r8 + 8*half, N = j
            Bout[(yb + 8 * half + r8) * TPW + x0 + j] = acc[r8];
    }
    __syncthreads();

    // ---------- Pass 2: vertical Gaussian (band matrix as A operand) ----------
    for (int t = wave; t < 75; t += 8) {
        int ch = t / 25;
        int r  = t - ch * 25;
        int ty = r / 5, tx = r - ty * 5;
        int y0 = (ty < 4) ? 2 + 16 * ty : 54;
        int x0 = (tx < 4) ? 2 + 16 * tx : 54;
        const float* Bin = &sB[ch * LSLAB];
        v8f acc = {};
#pragma unroll
        for (int s = 0; s < 5; ++s) {
            int row = y0 - 2 + 4 * s + 2 * half;     // <= 70 (+1 below -> 71)
            v2f b;
            b.x = Bin[row * TPW + x0 + j];
            b.y = Bin[(row + 1) * TPW + x0 + j];
            acc = __builtin_amdgcn_wmma_f32_16x16x4_f32(
                      false, band[s], false, b, (short)0, acc, false, false);
        }
        float* Aout = &sA[ch * LSLAB];               // blur overwrites input slab
#pragma unroll
        for (int r8 = 0; r8 < 8; ++r8)
            Aout[(y0 + 8 * half + r8) * TPW + x0 + j] = acc[r8];
    }
    __syncthreads();

    // ---------- Pass 3: Sobel -> grad_mag (sum of per-channel L2), sum gx, sum gy ----------
    // grad region: local [3..68]^2 (outputs +/-1 NMS halo).
    // Sobel reads of blur are zeroed outside the image (reference zero-pads b);
    // grad itself is zeroed outside the image (reference zero-pads grad_mag for NMS).
    for (int p = tid; p < 66 * 66; p += 256) {
        int py = p / 66, px = p - py * 66;
        int y = 3 + py, x = 3 + px;
        int gyy = gy0 + y - 4, gxx = gx0 + x - 4;
        float ms = 0.f, gxs = 0.f, gys = 0.f;
        if (gyy >= 0 && gyy < HH && gxx >= 0 && gxx < WW) {
#pragma unroll
            for (int ch = 0; ch < 3; ++ch) {
                const float* B = &sA[ch * LSLAB];
                float b[3][3];
#pragma unroll
                for (int dy = -1; dy <= 1; ++dy)
#pragma unroll
                    for (int dx = -1; dx <= 1; ++dx) {
                        int yy = gyy + dy, xx = gxx + dx;
                        float v = 0.f;
                        if (yy >= 0 && yy < HH && xx >= 0 && xx < WW)
                            v = B[(y + dy) * TPW + (x + dx)];
                        b[dy + 1][dx + 1] = v;
                    }
                float gxv = (b[0][0] - b[0][2]) + 2.f * (b[1][0] - b[1][2]) + (b[2][0] - b[2][2]);
                float gyv = (b[0][0] + 2.f * b[0][1] + b[0][2]) - (b[2][0] + 2.f * b[2][1] + b[2][2]);
                ms  += sqrtf(gxv * gxv + gyv * gyv);
                gxs += gxv;
                gys += gyv;
            }
        }
        sB[0 * LSLAB + y * TPW + x] = ms;    // hblur no longer needed -> reuse sB
        sB[1 * LSLAB + y * TPW + x] = gxs;
        sB[2 * LSLAB + y * TPW + x] = gys;
    }
    __syncthreads();

    // ---------- Pass 4: orientation, NMS, thresholds, all six outputs ----------
    // Compass offsets (dy,dx)+1 packed as nibbles:
    // offy = {0,1,1,1,0,-1,-1,-1} -> 0x00012221 ; offx = {1,1,0,-1,-1,-1,0,1} -> 0x21000122
    const unsigned OY = 0x00012221u;
    const unsigned OX = 0x21000122u;
    for (int q = tid; q < 64 * 64; q += 256) {
        int oy = q >> 6, ox = q & 63;
        int ly = 4 + oy, lx = 4 + ox;
        size_t gi = (size_t)(gy0 + oy) * WW + (gx0 + ox);
        float m   = sB[0 * LSLAB + ly * TPW + lx];
        float gxs = sB[1 * LSLAB + ly * TPW + lx];
        float gys = sB[2 * LSLAB + ly * TPW + lx];
        // orient = round((atan2*180/PI + 180)/45)*45 ; module PI literal = 3.14159
        float ang = atan2f(gys, gxs) * (180.0f / 3.14159f) + 180.0f;  // [0,360]
        int k = (int)rintf(ang * (1.0f / 45.0f));                      // [0,8], half-even like jnp.round
        int ip = k & 7, in_ = (k + 4) & 7;
        int dyp = (int)((OY >> (4 * ip)) & 15u) - 1;
        int dxp = (int)((OX >> (4 * ip)) & 15u) - 1;
        int dyn = (int)((OY >> (4 * in_)) & 15u) - 1;
        int dxn = (int)((OX >> (4 * in_)) & 15u) - 1;
        float pos = m - sB[0 * LSLAB + (ly + dyp) * TPW + (lx + dxp)];
        float neg = m - sB[0 * LSLAB + (ly + dyn) * TPW + (lx + dxn)];
        float thin = (fminf(pos, neg) > 0.0f) ? m : 0.0f;

        out[0 * (size_t)HWSZ + gi] = sA[0 * LSLAB + ly * TPW + lx];  // blurred ch0
        out[1 * (size_t)HWSZ + gi] = sA[1 * LSLAB + ly * TPW + lx];  // blurred ch1
        out[2 * (size_t)HWSZ + gi] = sA[2 * LSLAB + ly * TPW + lx];  // blurred ch2
        out[3 * (size_t)HWSZ + gi] = m;                              // grad_mag
        out[4 * (size_t)HWSZ + gi] = 45.0f * (float)k;               // orient
        out[5 * (size_t)HWSZ + gi] = thin;                           // thin_edges
        out[6 * (size_t)HWSZ + gi] = (thin < 10.0f) ? 0.0f : thin;   // thresholded
        out[7 * (size_t)HWSZ + gi] = (m    < 10.0f) ? 0.0f : m;      // early_threshold
    }
}

extern "C" void kernel_launch(void* const* d_in, const int* in_sizes, int n_in,
                              void* d_out, int out_size, void* d_ws, size_t ws_size,
                              hipStream_t stream) {
    const float* img   = (const float*)d_in[0];   // [1,3,2048,2048] f32
    const float* gauss = (const float*)d_in[1];   // [5] f32
    // d_in[2] (sobel) and d_in[3] (dir_filters) are fixed constants, hardcoded above.
    float* out = (float*)d_out;                   // 8*H*W f32, concatenated outputs
    (void)in_sizes; (void)n_in; (void)out_size; (void)d_ws; (void)ws_size;

    dim3 grid(WW / 64, HH / 64);                  // 32 x 32 tiles
    canny_fused_kernel<<<grid, 256, 0, stream>>>(img, gauss, out);
}